// DTCWTInverse_42709154791433
// MI455X (gfx1250) — compile-verified
//
#include <hip/hip_runtime.h>

// ---------------------------------------------------------------------------
// Inverse DTCWT for gfx1250 (MI455X).
// Final-level separable filtering = banded-matrix products on the matrix pipe
// (V_WMMA_F32_16X16X4_F32, full f32 precision). Operands staged through LDS
// with GLOBAL_LOAD_ASYNC_TO_LDS_B32 (ASYNCcnt) when the toolchain exposes it.
// Upsampling levels are VALU polyphase kernels with immediate-folded taps.
// ---------------------------------------------------------------------------

typedef __attribute__((ext_vector_type(2))) float v2f;
typedef __attribute__((ext_vector_type(8))) float v8f;

#define NIMG 128            // N*C = 8*16 images
#define CDIV(a,b) (((a)+(b)-1)/(b))

// ---- async global->LDS support probe --------------------------------------
#ifdef __gfx1250__
#if __has_builtin(__builtin_amdgcn_global_load_async_to_lds_b32)
#define HAVE_ASYNC_LDS 1
#endif
#endif
#ifndef HAVE_ASYNC_LDS
#define HAVE_ASYNC_LDS 0
#endif

#if HAVE_ASYNC_LDS
typedef __attribute__((address_space(1))) int g1int;   // global (AS1) int*
typedef __attribute__((address_space(3))) int l3int;   // LDS (AS3) int*
__device__ __forceinline__ void async_g2l(const float* g, float* l) {
  __builtin_amdgcn_global_load_async_to_lds_b32((g1int*)g, (l3int*)l, 0, 0);
}
#endif

__device__ __forceinline__ void wait_async0() {
#if HAVE_ASYNC_LDS
#if __has_builtin(__builtin_amdgcn_s_wait_asynccnt)
  __builtin_amdgcn_s_wait_asynccnt(0);
#else
  asm volatile("s_wait_asynccnt 0" ::: "memory");
#endif
#endif
}

// ---------------------------------------------------------------------------
__device__ __forceinline__ int refl(int q, int n) {
  // symmetric padding index (single reflection is enough: pad <= 5, n >= 32)
  if (q < 0)  q = -1 - q;
  if (q >= n) q = 2 * n - 1 - q;
  return q;
}

// Level-0 banded-matrix coefficients over a 20-row window starting at padded
// index (tile_base - 2). Filters are symmetric -> selects between immediates.
__device__ __forceinline__ float cf3(int m, int j) {
  const int d = j - m - 1;
  return (d == 1) ? 0.70710678118655f
       : ((d == 0) || (d == 2)) ? 0.35355339059327f
       : 0.0f;
}
__device__ __forceinline__ float cf5(int m, int j) {
  const int d = j - m;
  return (d == 2) ? 0.53033008588991f
       : ((d == 1) || (d == 3)) ? -0.17677669529664f
       : ((d == 0) || (d == 4)) ? -0.08838834764832f
       : 0.0f;
}

// 10-tap quarter-shift prototype as a constexpr table (folds to immediates).
__device__ __forceinline__ constexpr float g0aC(int i) {
  return (i == 0) ?  0.03516384f
       : (i == 2) ? -0.08832942f
       : (i == 3) ?  0.23389032f
       : (i == 4) ?  0.76027237f
       : (i == 5) ?  0.5875183f
       : (i == 7) ? -0.11430184f
       : 0.0f;
}

// Polyphase taps for colifilt/rowifilt, output index R = 4t+p.
//  low path:  ha = g0b (ha[q]=G0A[9-q]),        hb = g0a (hb[q]=G0A[q])
//  high path: ha = g1b (ha[q]=-(-1)^q G0A[q]),  hb = g1a (hb[q]=(-1)^q G0A[9-q])
//  p even -> ha[(p>=2?9:8)-2j]; p odd -> hb[(p>=2?9:8)-2j]
// j is a compile-time constant after unrolling -> pure immediate selects on p.
__device__ __forceinline__ void iftaps(int p, int j, float& tl, float& th) {
  const int qe = 8 - 2 * j;          // q when p < 2
  const int qo = 9 - 2 * j;          // q when p >= 2
  const float lo_a_e = g0aC(9 - qe), lo_a_o = g0aC(9 - qo);
  const float lo_b_e = g0aC(qe),     lo_b_o = g0aC(qo);
  const float hi_a_e = (qe & 1) ?  g0aC(qe)     : -g0aC(qe);
  const float hi_a_o = (qo & 1) ?  g0aC(qo)     : -g0aC(qo);
  const float hi_b_e = (qe & 1) ? -g0aC(9 - qe) :  g0aC(9 - qe);
  const float hi_b_o = (qo & 1) ? -g0aC(9 - qo) :  g0aC(9 - qo);
  const bool odd = (p & 1) != 0;
  const bool hiq = p >= 2;
  tl = odd ? (hiq ? lo_b_o : lo_b_e) : (hiq ? lo_a_o : lo_a_e);
  th = odd ? (hiq ? hi_b_o : hi_b_e) : (hiq ? hi_a_o : hi_a_e);
}

// ---------------------------------------------------------------------------
// c2q: pair of complex subbands -> real quad image (rows & cols doubled).
// yh layout: (img, 6, h, w, 2). lh = pair(0,5), hl = pair(2,3), hh = pair(1,4).
// ---------------------------------------------------------------------------
__global__ void c2q_kernel(const float* __restrict__ yh, int h, int w,
                           float* __restrict__ lh, float* __restrict__ hl,
                           float* __restrict__ hh) {
  int tid = blockIdx.x * blockDim.x + threadIdx.x;
  int hw = h * w;
  if (tid >= NIMG * hw) return;
  int b = tid % w;
  int a = (tid / w) % h;
  size_t img = (size_t)(tid / hw);
  size_t sb  = (size_t)hw * 2;
  const float* base = yh + img * 6 * sb + ((size_t)a * w + b) * 2;
  int h2 = 2 * h, w2 = 2 * w;
  const float s = 0.70710678118654752440f;
  size_t rb = (img * (size_t)h2 + 2 * a) * (size_t)w2 + 2 * b;

  {
    float w1r = base[0 * sb], w1i = base[0 * sb + 1];
    float w2r = base[5 * sb], w2i = base[5 * sb + 1];
    lh[rb]          = (w1r + w2r) * s;
    lh[rb + 1]      = (w1i + w2i) * s;
    lh[rb + w2]     = (w1i - w2i) * s;
    lh[rb + w2 + 1] = (w2r - w1r) * s;
  }
  {
    float w1r = base[2 * sb], w1i = base[2 * sb + 1];
    float w2r = base[3 * sb], w2i = base[3 * sb + 1];
    hl[rb]          = (w1r + w2r) * s;
    hl[rb + 1]      = (w1i + w2i) * s;
    hl[rb + w2]     = (w1i - w2i) * s;
    hl[rb + w2 + 1] = (w2r - w1r) * s;
  }
  {
    float w1r = base[1 * sb], w1i = base[1 * sb + 1];
    float w2r = base[4 * sb], w2i = base[4 * sb + 1];
    hh[rb]          = (w1r + w2r) * s;
    hh[rb + 1]      = (w1i + w2i) * s;
    hh[rb + w2]     = (w1i - w2i) * s;
    hh[rb + w2 + 1] = (w2r - w1r) * s;
  }
}

// ---------------------------------------------------------------------------
// out(2h x w) = colifilt(la, g0b, g0a, false) + colifilt(hb, g1b, g1a, true)
// ---------------------------------------------------------------------------
__global__ void upcol_kernel(const float* __restrict__ la,
                             const float* __restrict__ hb,
                             int hIn, int w, float* __restrict__ out) {
  int Ho  = 2 * hIn;
  int tid = blockIdx.x * blockDim.x + threadIdx.x;
  if (tid >= NIMG * Ho * w) return;
  int cc = tid % w;
  int R  = (tid / w) % Ho;
  size_t img = (size_t)(tid / (w * Ho));
  int t = R >> 2, p = R & 3;
  int oL = (p & 1) ? 2 : 1;
  int oH = 3 - oL;
  const float* A = la + img * (size_t)hIn * w + cc;
  const float* B = hb + img * (size_t)hIn * w + cc;
  float acc = 0.0f;
#pragma unroll
  for (int j = 0; j < 5; ++j) {
    float tl, th;
    iftaps(p, j, tl, th);
    int ra = refl(oL + 2 * (t + j) - 5, hIn);
    int rb = refl(oH + 2 * (t + j) - 5, hIn);
    acc = fmaf(tl, A[(size_t)ra * w], acc);
    acc = fmaf(th, B[(size_t)rb * w], acc);
  }
  out[tid] = acc;
}

// ---------------------------------------------------------------------------
// out(h x 2w) = rowifilt(y1, g0b, g0a, false) + rowifilt(y2, g1b, g1a, true)
// ---------------------------------------------------------------------------
__global__ void uprow_kernel(const float* __restrict__ y1,
                             const float* __restrict__ y2,
                             int h, int w, float* __restrict__ out) {
  int Wo  = 2 * w;
  int tid = blockIdx.x * blockDim.x + threadIdx.x;
  if (tid >= NIMG * h * Wo) return;
  int Cc = tid % Wo;
  int r  = (tid / Wo) % h;
  size_t img = (size_t)(tid / (h * Wo));
  int t = Cc >> 2, p = Cc & 3;
  int oL = (p & 1) ? 2 : 1;
  int oH = 3 - oL;
  const float* A = y1 + (img * h + r) * (size_t)w;
  const float* B = y2 + (img * h + r) * (size_t)w;
  float acc = 0.0f;
#pragma unroll
  for (int j = 0; j < 5; ++j) {
    float tl, th;
    iftaps(p, j, tl, th);
    int ca = refl(oL + 2 * (t + j) - 5, w);
    int cb = refl(oH + 2 * (t + j) - 5, w);
    acc = fmaf(tl, A[ca], acc);
    acc = fmaf(th, B[cb], acc);
  }
  out[tid] = acc;
}

// ---------------------------------------------------------------------------
// Level-0 column pass on the matrix pipe. 4 waves / block; each block handles
// a 16-row x 64-col strip of y1 or y2. Window (20 rows x 64 cols, reflected)
// staged into LDS (async), then per wave:
//   Y(16x16) = M3(16x20)*S0win(20x16) + M5(16x20)*S1win(20x16)
// as 5 K=4 chunks -> 10 x V_WMMA_F32_16X16X4_F32.
// Pitch 72: half-waves read rows kb and kb+2 -> bank offset 16 -> no conflicts.
// ---------------------------------------------------------------------------
#define CP_PITCH 72
__global__ void __launch_bounds__(128)
final_col_wmma(const float* __restrict__ zsrc, const float* __restrict__ lhsrc,
               const float* __restrict__ hlsrc, const float* __restrict__ hhsrc,
               float* __restrict__ y1out, float* __restrict__ y2out,
               int H, int W) {
  __shared__ float s0[20 * CP_PITCH];
  __shared__ float s1[20 * CP_PITCH];
  const int tid  = threadIdx.x;
  const int lane = tid & 31;
  const int wv   = tid >> 5;            // wave 0..3 -> 16-col sub-tile
  const int n    = lane & 15;           // A-matrix M index == B-matrix N index
  const int hi   = lane >> 4;           // half-wave -> K pair {0,1} vs {2,3}
  const int cs   = blockIdx.x * 64;
  const int r0   = blockIdx.y * 16;
  const size_t img = (size_t)(blockIdx.z % NIMG);
  const int which  = (int)(blockIdx.z / NIMG);
  const float* S0 = (which ? hlsrc : zsrc)  + img * (size_t)H * W;
  const float* S1 = (which ? hhsrc : lhsrc) + img * (size_t)H * W;
  float* dst      = (which ? y2out : y1out) + img * (size_t)H * W;

  // stage 20x64 reflected row window, coalesced (10 iterations, uniform)
  for (int i = tid; i < 20 * 64; i += 128) {
    int rr = i >> 6, cc = i & 63;
    size_t gidx = (size_t)refl(r0 - 2 + rr, H) * W + cs + cc;
    int l = rr * CP_PITCH + cc;
#if HAVE_ASYNC_LDS
    async_g2l(S0 + gidx, &s0[l]);
    async_g2l(S1 + gidx, &s1[l]);
#else
    s0[l] = S0[gidx];
    s1[l] = S1[gidx];
#endif
  }
  wait_async0();
  __syncthreads();

  const int cw = wv * 16;
  v8f acc = {};
#pragma unroll
  for (int k0 = 0; k0 < 20; k0 += 4) {
    const int kb = k0 + 2 * hi;
    v2f a3, a5, b0, b1;
    a3.x = cf3(n, kb); a3.y = cf3(n, kb + 1);
    a5.x = cf5(n, kb); a5.y = cf5(n, kb + 1);
    b0.x = s0[kb * CP_PITCH + cw + n];
    b0.y = s0[(kb + 1) * CP_PITCH + cw + n];
    b1.x = s1[kb * CP_PITCH + cw + n];
    b1.y = s1[(kb + 1) * CP_PITCH + cw + n];
    acc = __builtin_amdgcn_wmma_f32_16x16x4_f32(false, a3, false, b0,
                                                (short)0, acc, false, false);
    acc = __builtin_amdgcn_wmma_f32_16x16x4_f32(false, a5, false, b1,
                                                (short)0, acc, false, false);
  }
  // D layout: VGPR j -> row (j + 8*hi), col = n
#pragma unroll
  for (int j = 0; j < 8; ++j)
    dst[(size_t)(r0 + j + 8 * hi) * W + cs + cw + n] = acc[j];
}

// ---------------------------------------------------------------------------
// Level-0 row pass: Out(16x16) = Y1win(16x20)*M3^T + Y2win(16x20)*M5^T.
// 4 waves / block over a 16x64 strip; 16x72 reflected col window in LDS.
// Pitch 73: lane n reads word 73*n + c -> bank 9n + c (9n is a permutation
// mod 64; with half-wave offset 2, n=0,1 share banks with n=14,15: 14*73+2 = 1024).
// ---------------------------------------------------------------------------
#define RP_PITCH 73
__global__ void __launch_bounds__(128)
final_row_wmma(const float* __restrict__ y1, const float* __restrict__ y2,
               float* __restrict__ out, int H, int W) {
  __shared__ float sh1[16 * RP_PITCH];
  __shared__ float sh2[16 * RP_PITCH];
  const int tid  = threadIdx.x;
  const int lane = tid & 31;
  const int wv   = tid >> 5;
  const int n    = lane & 15;
  const int hi   = lane >> 4;
  const int cs   = blockIdx.x * 64;
  const int r0   = blockIdx.y * 16;
  const size_t img = (size_t)blockIdx.z;
  const float* A1 = y1 + img * (size_t)H * W;
  const float* A2 = y2 + img * (size_t)H * W;

  // stage 16x72 reflected col window, coalesced (9 iterations, uniform)
  for (int i = tid; i < 16 * 72; i += 128) {
    int rr = i / 72, jj = i - rr * 72;
    size_t gidx = (size_t)(r0 + rr) * W + refl(cs - 2 + jj, W);
    int l = rr * RP_PITCH + jj;
#if HAVE_ASYNC_LDS
    async_g2l(A1 + gidx, &sh1[l]);
    async_g2l(A2 + gidx, &sh2[l]);
#else
    sh1[l] = A1[gidx];
    sh2[l] = A2[gidx];
#endif
  }
  wait_async0();
  __syncthreads();

  const int cw = wv * 16;               // window base for this wave's sub-tile
  v8f acc = {};
#pragma unroll
  for (int k0 = 0; k0 < 20; k0 += 4) {
    const int kb = k0 + 2 * hi;
    v2f aa, ab, b3, b5;
    const float* r1 = &sh1[n * RP_PITCH + cw + kb];
    const float* r2 = &sh2[n * RP_PITCH + cw + kb];
    aa.x = r1[0]; aa.y = r1[1];
    ab.x = r2[0]; ab.y = r2[1];
    b3.x = cf3(n, kb); b3.y = cf3(n, kb + 1);
    b5.x = cf5(n, kb); b5.y = cf5(n, kb + 1);
    acc = __builtin_amdgcn_wmma_f32_16x16x4_f32(false, aa, false, b3,
                                                (short)0, acc, false, false);
    acc = __builtin_amdgcn_wmma_f32_16x16x4_f32(false, ab, false, b5,
                                                (short)0, acc, false, false);
  }
  float* dst = out + img * (size_t)H * W;
#pragma unroll
  for (int j = 0; j < 8; ++j)
    dst[(size_t)(r0 + j + 8 * hi) * W + cs + cw + n] = acc[j];
}

// ---------------------------------------------------------------------------
extern "C" void kernel_launch(void* const* d_in, const int* in_sizes, int n_in,
                              void* d_out, int out_size, void* d_ws, size_t ws_size,
                              hipStream_t stream) {
  (void)in_sizes; (void)n_in; (void)out_size; (void)ws_size;
  const float* yl  = (const float*)d_in[0];
  const float* yh0 = (const float*)d_in[1];
  const float* yh1 = (const float*)d_in[2];
  const float* yh2 = (const float*)d_in[3];

  float* ws = (float*)d_ws;
  const size_t I  = NIMG;
  const size_t B0 = I * 256 * 256;
  const size_t B1 = I * 128 * 128;
  const size_t B2 = I * 64 * 64;

  float* bands0 = ws;                 // 3*B0  (lh0, hl0, hh0)
  float* bandsS = ws + 3 * B0;        // 3*B1  (level-2 then level-1 bands)
  float* z1     = bandsS + 3 * B1;    // B1
  float* z2     = z1 + B1;            // B0
  float* t1     = z2 + B0;            // B0  (y1 staging)
  float* t2     = t1 + B0;            // B0  (y2 staging)

  const int TB = 256;

  // ---- level 2: yl (64x64) + yh2 bands (64x64) -> z1 (128x128) ----
  { int tot = (int)(I * 32 * 32);
    c2q_kernel<<<CDIV(tot, TB), TB, 0, stream>>>(yh2, 32, 32,
                                                 bandsS, bandsS + B2, bandsS + 2 * B2); }
  { int tot = (int)(I * 128 * 64);
    upcol_kernel<<<CDIV(tot, TB), TB, 0, stream>>>(yl, bandsS, 64, 64, t1);
    upcol_kernel<<<CDIV(tot, TB), TB, 0, stream>>>(bandsS + B2, bandsS + 2 * B2,
                                                   64, 64, t2); }
  { int tot = (int)(I * 128 * 128);
    uprow_kernel<<<CDIV(tot, TB), TB, 0, stream>>>(t1, t2, 128, 64, z1); }

  // ---- level 1: z1 (128x128) + yh1 bands (128x128) -> z2 (256x256) ----
  { int tot = (int)(I * 64 * 64);
    c2q_kernel<<<CDIV(tot, TB), TB, 0, stream>>>(yh1, 64, 64,
                                                 bandsS, bandsS + B1, bandsS + 2 * B1); }
  { int tot = (int)(I * 256 * 128);
    upcol_kernel<<<CDIV(tot, TB), TB, 0, stream>>>(z1, bandsS, 128, 128, t1);
    upcol_kernel<<<CDIV(tot, TB), TB, 0, stream>>>(bandsS + B1, bandsS + 2 * B1,
                                                   128, 128, t2); }
  { int tot = (int)(I * 256 * 256);
    uprow_kernel<<<CDIV(tot, TB), TB, 0, stream>>>(t1, t2, 256, 128, z2); }

  // ---- level 0: z2 + yh0 bands -> out (256x256), on the matrix pipe ----
  { int tot = (int)(I * 128 * 128);
    c2q_kernel<<<CDIV(tot, TB), TB, 0, stream>>>(yh0, 128, 128,
                                                 bands0, bands0 + B0, bands0 + 2 * B0); }
  { dim3 g(256 / 64, 256 / 16, (unsigned)(2 * I));
    final_col_wmma<<<g, 128, 0, stream>>>(z2, bands0, bands0 + B0, bands0 + 2 * B0,
                                          t1, t2, 256, 256); }
  { dim3 g(256 / 64, 256 / 16, (unsigned)I);
    final_row_wmma<<<g, 128, 0, stream>>>(t1, t2, (float*)d_out, 256, 256); }
}